// ChunkData_23106924053186
// MI455X (gfx1250) — compile-verified
//
#include <hip/hip_runtime.h>

// ChunkData: x[n,f,c] = mixed[n+c, f]  (N=16320, F=257, C=64), y = clean[64:]
// Store-bandwidth-bound (~1.09 GB writes @ 23.3 TB/s ~= 47us floor).
// Strategy: TDM (tensor_load_to_lds) stages 127x64 input tiles into LDS with
// a 1-dword pad per 64-dword row (stride 65 -> conflict-free transpose reads),
// then waves emit fully coalesced 128-bit non-temporal stores along c.

#define T_DIM   16384
#define F_DIM   257
#define CHUNK   64
#define N_OUT   (T_DIM - CHUNK)   // 16320
#define FT      64                // f columns per tile
#define NT      64                // n values per tile
#define RROWS   127               // NT + CHUNK - 1 input rows per tile

typedef __attribute__((ext_vector_type(4))) unsigned int u32x4;
typedef __attribute__((ext_vector_type(8))) int          i32x8;
typedef __attribute__((ext_vector_type(4))) int          i32x4;
typedef __attribute__((ext_vector_type(4))) float        f32x4;

__global__ __launch_bounds__(256)
void chunk_x_kernel(const float* __restrict__ in, float* __restrict__ out) {
    // 127 rows x 65 dwords (TDM pads 1 dword per 64-dword row) = 33,020 B
    __shared__ float tile[RROWS * (FT + 1)];

    const int fTile = blockIdx.x;          // 0..4
    const int nTile = blockIdx.y;          // 0..254
    const int f0    = fTile * FT;
    const int n0    = nTile * NT;
    const int ftAct = (F_DIM - f0) < FT ? (F_DIM - f0) : FT;   // 64 or 1
    const int ntAct = (N_OUT - n0) < NT ? (N_OUT - n0) : NT;   // always 64
    const bool pad  = (ftAct == FT);
    const int  strd = pad ? (FT + 1) : ftAct;                  // LDS row stride

    // ---- TDM load: rows [n0, n0+127) x cols [f0, f0+ftAct) -> LDS ----
    // Issue from wave 0 only (tensor ops ignore EXEC; keep it one wave).
    if (__builtin_amdgcn_readfirstlane((int)(threadIdx.x >> 5)) == 0) {
        // Flat LDS address: addr[31:0] is the LDS byte offset (ISA 10.2).
        unsigned lds_base = (unsigned)(unsigned long long)(void*)tile;
        unsigned long long ga = (unsigned long long)(const void*)in
                              + 4ull * ((unsigned long long)n0 * F_DIM + f0);

        // D# group 0 (128b): count=1 | lds_addr | global_addr[56:0] | type=2
        u32x4 g0;
        g0.x = 1u;
        g0.y = lds_base;
        g0.z = (unsigned)ga;
        g0.w = ((unsigned)(ga >> 32) & 0x1FFFFFFu) | (2u << 30);

        // D# group 1 (256b)
        i32x8 g1;
        // [17:16] data_size=2 (4B); [20] pad_en; [24:22] pad_interval=5 (64 dw);
        // [31:25] pad_amount=0 (1 dw)
        g1[0] = (int)((2u << 16) | (pad ? ((1u << 20) | (5u << 22)) : 0u));
        g1[1] = (int)(((unsigned)F_DIM & 0xFFFFu) << 16);   // tensor_dim0 lo16
        g1[2] = (int)(((unsigned)T_DIM & 0xFFFFu) << 16);   // dim0 hi=0 | tensor_dim1 lo16
        g1[3] = (int)(((unsigned)ftAct) << 16);             // dim1 hi=0 | tile_dim0
        g1[4] = RROWS;                                      // tile_dim1 | tile_dim2=0
        g1[5] = F_DIM;                                      // tensor_dim0_stride lo32
        g1[6] = 0;                                          // stride0 hi | stride1 lo16
        g1[7] = 0;

        i32x4 gz = {0, 0, 0, 0};
#if defined(__clang_major__) && (__clang_major__ >= 23)
        i32x8 gz8 = {0, 0, 0, 0, 0, 0, 0, 0};
        __builtin_amdgcn_tensor_load_to_lds(g0, g1, gz, gz, gz8, 0);
#else
        __builtin_amdgcn_tensor_load_to_lds(g0, g1, gz, gz, 0);
#endif
        __builtin_amdgcn_s_wait_tensorcnt(0);
    }
    __syncthreads();

    // ---- Transpose out of LDS, coalesced float4 stores along c ----
    // lane tid: c = 4*(tid&15); (tid>>4) walks f -> lanes 0..31 cover 512 B
    // contiguous output. LDS reads: bank (base + 4*(tid&15)) % 64, conflict-free.
    const int tid   = (int)threadIdx.x;
    const int c     = (tid & 15) * 4;
    const int fBase = tid >> 4;

    for (int nl = 0; nl < ntAct; ++nl) {
        float* __restrict__ outRow = out + ((long long)(n0 + nl) * F_DIM) * CHUNK;
        const int ldsRow = (nl + c) * strd;
        for (int fl = fBase; fl < ftAct; fl += 16) {
            const int b = ldsRow + fl;
            f32x4 v;
            v.x = tile[b];
            v.y = tile[b + strd];
            v.z = tile[b + 2 * strd];
            v.w = tile[b + 3 * strd];
            __builtin_nontemporal_store(v,
                (f32x4*)(outRow + (long long)(f0 + fl) * CHUNK + c));
        }
    }
}

__global__ __launch_bounds__(256)
void copy_y_kernel(const float* __restrict__ in, float* __restrict__ out, int n4) {
    int i = (int)(blockIdx.x * blockDim.x + threadIdx.x);
    if (i < n4) {
        f32x4 v = __builtin_nontemporal_load((const f32x4*)in + i);
        __builtin_nontemporal_store(v, (f32x4*)out + i);
    }
}

extern "C" void kernel_launch(void* const* d_in, const int* in_sizes, int n_in,
                              void* d_out, int out_size, void* d_ws, size_t ws_size,
                              hipStream_t stream) {
    const float* mixed = (const float*)d_in[0];   // (16384, 257) f32
    const float* clean = (const float*)d_in[1];   // (16384, 257) f32
    float* out = (float*)d_out;                   // x (16320*257*64) ++ y (16320*257)

    // x: 5 f-tiles x 255 n-tiles
    dim3 grid((F_DIM + FT - 1) / FT, N_OUT / NT);
    chunk_x_kernel<<<grid, 256, 0, stream>>>(mixed, out);

    // y: flat float4 copy of clean[64:, :]
    const int yElems = N_OUT * F_DIM;             // 4,194,240 (divisible by 4)
    const int n4 = yElems / 4;
    float* yOut = out + (long long)N_OUT * F_DIM * CHUNK;
    copy_y_kernel<<<(n4 + 255) / 256, 256, 0, stream>>>(
        clean + CHUNK * F_DIM, yOut, n4);
}